// GraphNN_Model_14259291422821
// MI455X (gfx1250) — compile-verified
//
#include <hip/hip_runtime.h>

typedef float v2f __attribute__((ext_vector_type(2)));
typedef float v8f __attribute__((ext_vector_type(8)));

#define K_DIM 128  // inner dim of every GEMM in this model

// ---------------------------------------------------------------------------
// Degree computation: deg starts at 1.0 (self loop), atomic +1 per in-edge,
// then in-place rsqrt -> deg_inv_sqrt. deg >= 1 always, so no zero guard.
// ---------------------------------------------------------------------------
__global__ __launch_bounds__(256) void init_deg(float* deg, int n) {
  int i = blockIdx.x * blockDim.x + threadIdx.x;
  if (i < n) deg[i] = 1.0f;
}

__global__ __launch_bounds__(256) void deg_accum(const long long* __restrict__ dst,
                                                 float* __restrict__ deg, int E) {
  int e = blockIdx.x * blockDim.x + threadIdx.x;
  if (e < E) atomicAdd(&deg[(int)dst[e]], 1.0f);
}

__global__ __launch_bounds__(256) void deg_rsqrt(float* deg, int n) {
  int i = blockIdx.x * blockDim.x + threadIdx.x;
  if (i < n) deg[i] = rsqrtf(deg[i]);
}

// ---------------------------------------------------------------------------
// agg[i,f] = dis[i]^2 * h[i,f] + b[f]   (self-loop message + bias; also
// serves as the zero-init of the aggregation buffer). F is fixed at 128.
// ---------------------------------------------------------------------------
__global__ __launch_bounds__(256) void self_loop_bias(const float* __restrict__ h,
                                                      const float* __restrict__ dis,
                                                      const float* __restrict__ b,
                                                      float* __restrict__ agg,
                                                      int total /* = N*128 */) {
  int idx = blockIdx.x * blockDim.x + threadIdx.x;
  if (idx >= total) return;
  int i = idx >> 7;       // node
  int f = idx & 127;      // feature
  float s = dis[i];
  agg[idx] = s * s * h[idx] + b[f];
}

// ---------------------------------------------------------------------------
// Edge aggregation: one wave32 per edge. Each lane handles 4 consecutive
// features (32 lanes * 4 = 128). Gather h[src] (float4, L2-resident),
// scale by dis[src]*dis[dst], global_atomic_add_f32 into agg[dst].
// ---------------------------------------------------------------------------
__global__ __launch_bounds__(256) void edge_agg(const long long* __restrict__ src,
                                                const long long* __restrict__ dst,
                                                const float* __restrict__ dis,
                                                const float* __restrict__ h,
                                                float* __restrict__ agg, int E) {
  int e = blockIdx.x * 8 + (threadIdx.x >> 5);
  if (e >= E) return;
  int lane = threadIdx.x & 31;
  int s = (int)src[e];
  int d = (int)dst[e];
  float norm = dis[s] * dis[d];
  float4 v = *(const float4*)(h + (size_t)s * 128 + lane * 4);
  float* out = agg + (size_t)d * 128 + lane * 4;
  atomicAdd(out + 0, norm * v.x);
  atomicAdd(out + 1, norm * v.y);
  atomicAdd(out + 2, norm * v.z);
  atomicAdd(out + 3, norm * v.w);
}

// ---------------------------------------------------------------------------
// FP32 WMMA GEMM:  C[M,NOUT] = act(A[M,128]) @ W[128,NOUT] (+ bias)
// One wave computes one 16x16 tile via 32x v_wmma_f32_16x16x4_f32.
// Block = (NOUT/16) waves, one tile-row of M per block. M % 16 == 0.
//
// VGPR layouts (CDNA5 ISA):
//   A 16x4 : lane = (k>=2)*16 + m, vgpr = k&1   -> per-lane float2 at even k
//   B 4x16 : lane = (k>=2)*16 + n, vgpr = k&1
//   C 16x16: lane l, vgpr v -> (m = v + 8*(l>=16), n = l&15)
// ---------------------------------------------------------------------------
template <int NOUT, bool RELU_A, bool HAS_BIAS>
__global__ __launch_bounds__(32 * (NOUT / 16)) void gemm_wmma_f32(
    const float* __restrict__ A, const float* __restrict__ W,
    const float* __restrict__ bias, float* __restrict__ C) {
  const int lane  = threadIdx.x & 31;
  const int tn    = threadIdx.x >> 5;    // N-tile owned by this wave
  const int tm    = blockIdx.x;          // M-tile owned by this block
  const int mn    = lane & 15;           // row (for A) / col (for B) index
  const int khalf = (lane >> 4) << 1;    // 0 or 2: K sub-offset per lane half

  const float* Arow = A + (size_t)(tm * 16 + mn) * K_DIM;   // A row of this lane
  const float* Wcol = W + tn * 16 + mn;                     // B col of this lane

  v8f c = {};
#pragma unroll
  for (int k0 = 0; k0 < K_DIM; k0 += 4) {
    const int kb = k0 + khalf;
    float2 av = *(const float2*)(Arow + kb);   // consecutive K -> 8B load
    v2f a, b;
    a.x = RELU_A ? fmaxf(av.x, 0.0f) : av.x;
    a.y = RELU_A ? fmaxf(av.y, 0.0f) : av.y;
    b.x = Wcol[(size_t)kb * NOUT];
    b.y = Wcol[(size_t)(kb + 1) * NOUT];
    c = __builtin_amdgcn_wmma_f32_16x16x4_f32(
        /*neg_a=*/false, a, /*neg_b=*/false, b,
        /*c_mod=*/(short)0, c, /*reuse_a=*/false, /*reuse_b=*/false);
  }

  const int mbase = tm * 16 + ((lane >> 4) << 3);  // +0 or +8
  float* Crow = C + (size_t)mbase * NOUT + tn * 16 + mn;
  const float badd = HAS_BIAS ? bias[tn * 16 + mn] : 0.0f;
#pragma unroll
  for (int v = 0; v < 8; ++v) {
    Crow[(size_t)v * NOUT] = c[v] + badd;
  }
}

// ---------------------------------------------------------------------------
// Orchestration
// ---------------------------------------------------------------------------
extern "C" void kernel_launch(void* const* d_in, const int* in_sizes, int n_in,
                              void* d_out, int out_size, void* d_ws, size_t ws_size,
                              hipStream_t stream) {
  const float*     x   = (const float*)d_in[0];
  const long long* ei  = (const long long*)d_in[1];  // int64 per reference
  const float*     W1  = (const float*)d_in[2];
  const float*     b1  = (const float*)d_in[3];
  const float*     W2  = (const float*)d_in[4];
  const float*     b2  = (const float*)d_in[5];
  const float*     Wf  = (const float*)d_in[6];
  const float*     bf  = (const float*)d_in[7];
  float*           out = (float*)d_out;

  const int N = in_sizes[0] / 128;  // 100000 (divisible by 16)
  const int E = in_sizes[1] / 2;    // 1600000
  const long long* src = ei;        // edge_index[0]
  const long long* dst = ei + E;    // edge_index[1]

  // Workspace layout (floats): dis[N] | h[N*128] | agg[N*128]  (~102.8 MB)
  float* dis = (float*)d_ws;
  float* h   = dis + N;
  float* agg = h + (size_t)N * 128;

  const int T = 256;
  const int total = N * 128;

  // deg_inv_sqrt
  init_deg<<<(N + T - 1) / T, T, 0, stream>>>(dis, N);
  deg_accum<<<(E + T - 1) / T, T, 0, stream>>>(dst, dis, E);
  deg_rsqrt<<<(N + T - 1) / T, T, 0, stream>>>(dis, N);

  // Layer 1: h = x @ W1 ; agg = D^-1/2 (A+I) D^-1/2 h + b1   (pre-ReLU)
  gemm_wmma_f32<128, false, false><<<N / 16, 256, 0, stream>>>(x, W1, nullptr, h);
  self_loop_bias<<<(total + T - 1) / T, T, 0, stream>>>(h, dis, b1, agg, total);
  edge_agg<<<(E + 7) / 8, 256, 0, stream>>>(src, dst, dis, h, agg, E);

  // Layer 2: h = relu(agg) @ W2 ; agg = aggregate + b2       (pre-ReLU)
  gemm_wmma_f32<128, true, false><<<N / 16, 256, 0, stream>>>(agg, W2, nullptr, h);
  self_loop_bias<<<(total + T - 1) / T, T, 0, stream>>>(h, dis, b2, agg, total);
  edge_agg<<<(E + 7) / 8, 256, 0, stream>>>(src, dst, dis, h, agg, E);

  // Final: out = relu(agg) @ Wf + bf
  gemm_wmma_f32<64, true, true><<<N / 16, 128, 0, stream>>>(agg, Wf, bf, out);
}